// DSC_PO_29721173688901
// MI455X (gfx1250) — compile-verified
//
#include <hip/hip_runtime.h>

// ---------------------------------------------------------------------------
// Problem constants (from reference): N=512, MC=16, T=512, H=10, M2=10, B=64
// ---------------------------------------------------------------------------
#define NDIM 512
#define MCTL 16
#define TLEN 512
#define HSP  10
#define MSP  10
#define BATCH 64
#define CHUNK 32              // t = 32*q + r ; r in [0,32), q in [0,16)
#define NQ   (TLEN / CHUNK)   // 16

typedef float v2f __attribute__((ext_vector_type(2)));
typedef float v8f __attribute__((ext_vector_type(8)));

__device__ __forceinline__ v8f wmma_f32_16x16x4(v2f a, v2f b, v8f c) {
  // V_WMMA_F32_16X16X4_F32 : D(16x16 f32) = A(16x4 f32) * B(4x16 f32) + C
  return __builtin_amdgcn_wmma_f32_16x16x4_f32(false, a, false, b, (short)0, c,
                                               false, false);
}

// ---------------------------------------------------------------------------
// Generic fp32 WMMA GEMM: D = A @ B (+ Cadd), row-major, optional batching via
// blockIdx.y with per-operand element strides.
//   NT  : N-tiles per wave (16*NT output columns) -> NT independent acc chains
//   KC  : K-split chains per N-tile -> KC more independent chains, summed at
//         the end (latency hiding for the serial Horner/power stages).
//   LDB : compile-time B leading dimension -> all B-fragment loads use
//         immediate offsets off ONE per-lane pointer; two pointer increments
//         per K-step total. Loads are staged before the WMMA burst so the
//         scheduler can clause them and use partial s_wait_loadcnt.
// Fragment layouts follow CDNA5 ISA 7.12.2:
//   A 16x4 : lane L holds row L%16, K = 2*(L/16)+{0,1}  (v2f, contiguous b64)
//   B 4x16 : lane L holds col L%16, K = 2*(L/16)+{0,1}
//   C/D    : lane L, VGPR v -> (row v + 8*(L/16), col L%16)
// Requires: M%16==0, N%(16*NT)==0, K%(4*KC)==0.
// ---------------------------------------------------------------------------
template <int NT, int KC, int LDB>
__global__ __launch_bounds__(256) void wmma_gemm(
    const float* __restrict__ Ag, int lda, long long sA,
    const float* __restrict__ Bg, long long sB,
    const float* __restrict__ Cg, int ldc, long long sC,
    float* __restrict__ Dg, int ldd, long long sD,
    int M, int N, int K)
{
  const int bz = blockIdx.y;
  const float* Ab = Ag + (long long)bz * sA;
  const float* Bb = Bg + (long long)bz * sB;
  const float* Cb = Cg ? (Cg + (long long)bz * sC) : nullptr;
  float*       Db = Dg + (long long)bz * sD;

  const int wid = blockIdx.x * (blockDim.x >> 5) + (threadIdx.x >> 5);
  const int tilesN = N / (16 * NT);
  const int tilesM = M / 16;
  if (wid >= tilesM * tilesN) return;            // whole-wave uniform exit
  const int tm = wid / tilesN, tn = wid % tilesN;
  const int rowBase = tm * 16;
  const int colBase = tn * 16 * NT;

  const int lane = threadIdx.x & 31;
  const int r16  = lane & 15;
  const int kh   = lane >> 4;                    // 0 or 1

  v8f acc[NT][KC];
#pragma unroll
  for (int t = 0; t < NT; ++t) {
#pragma unroll
    for (int kc = 0; kc < KC; ++kc) {
      v8f z = {0.f, 0.f, 0.f, 0.f, 0.f, 0.f, 0.f, 0.f};
      acc[t][kc] = z;
    }
    if (Cb) {
      const int col = colBase + t * 16 + r16;
#pragma unroll
      for (int v = 0; v < 8; ++v)
        acc[t][0][v] = Cb[(long long)(rowBase + v + 8 * kh) * ldc + col];
    }
  }

  // Single per-lane pointers; every fragment element is an immediate offset.
  const float* Ap = Ab + (long long)(rowBase + r16) * lda + 2 * kh;
  const float* Bp = Bb + (long long)(2 * kh) * LDB + colBase + r16;

  for (int k0 = 0; k0 < K; k0 += 4 * KC) {
    v2f a[KC];
    v2f b[KC][NT];
#pragma unroll
    for (int kc = 0; kc < KC; ++kc) {
      a[kc] = *(const v2f*)(Ap + 4 * kc);        // b64, imm offset 16*kc
#pragma unroll
      for (int t = 0; t < NT; ++t) {
        b[kc][t].x = Bp[(4 * kc + 0) * LDB + t * 16];   // b32, imm offset
        b[kc][t].y = Bp[(4 * kc + 1) * LDB + t * 16];   // b32, imm offset
      }
    }
#pragma unroll
    for (int kc = 0; kc < KC; ++kc)
#pragma unroll
      for (int t = 0; t < NT; ++t)
        acc[t][kc] = wmma_f32_16x16x4(a[kc], b[kc][t], acc[t][kc]);
    Ap += 4 * KC;
    Bp += 4 * KC * LDB;
  }

#pragma unroll
  for (int t = 0; t < NT; ++t) {
    v8f r = acc[t][0];
#pragma unroll
    for (int kc = 1; kc < KC; ++kc) r = r + acc[t][kc];
    const int col = colBase + t * 16 + r16;
#pragma unroll
    for (int v = 0; v < 8; ++v)
      Db[(long long)(rowBase + v + 8 * kh) * ldd + col] = r[v];
  }
}

// ---------------------------------------------------------------------------
// Small supporting kernels (all trivial FLOP counts)
// ---------------------------------------------------------------------------

// s_m[h] = sum_l sigma_phi_m[h, l]
__global__ void sm_kernel(const float* __restrict__ spm, float* __restrict__ sm) {
  int h = threadIdx.x;
  if (h < HSP) {
    float s = 0.f;
    for (int l = 0; l < MSP; ++l) s += spm[h * MSP + l];
    sm[h] = s;
  }
}

// M0s[c, n] = sum_h M0[c, h, n] * s_m[h]
__global__ void m0s_kernel(const float* __restrict__ M0,
                           const float* __restrict__ sm,
                           float* __restrict__ M0s) {
  int i = blockIdx.x * blockDim.x + threadIdx.x;      // 16*512
  if (i >= MCTL * NDIM) return;
  int c = i >> 9, n = i & (NDIM - 1);
  float s = 0.f;
  for (int h = 0; h < HSP; ++h) s += M0[((long long)c * HSP + h) * NDIM + n] * sm[h];
  M0s[i] = s;
}

// R[c, i, n] = sum_j M_tensor[c, i, j, n] * s_m[j]
__global__ void r_kernel(const float* __restrict__ Mt,
                         const float* __restrict__ sm,
                         float* __restrict__ R) {
  int i = blockIdx.x * blockDim.x + threadIdx.x;      // 16*10*512
  if (i >= MCTL * HSP * NDIM) return;
  int n = i & (NDIM - 1);
  int ci = i >> 9;                                    // c*10 + i
  float s = 0.f;
  for (int j = 0; j < MSP; ++j)
    s += Mt[((long long)ci * MSP + j) * NDIM + n] * sm[j];
  R[i] = s;
}

// W_0 = B : Wcat[n, 0:16] = B[n, :]   (Wcat row stride 512)
__global__ void copyB_kernel(const float* __restrict__ Bin, float* __restrict__ Wcat) {
  int i = blockIdx.x * blockDim.x + threadIdx.x;      // 512*16
  if (i >= NDIM * MCTL) return;
  int n = i >> 4, m = i & 15;
  Wcat[(long long)n * TLEN + m] = Bin[i];
}

// Ucat[q, r*16+m, b] = u_hist_rev[b, 32q+r, m]
__global__ void packU_kernel(const float* __restrict__ u, float* __restrict__ Ucat) {
  int i = blockIdx.x * blockDim.x + threadIdx.x;      // 16*512*64
  if (i >= NQ * TLEN * BATCH) return;
  int b  = i & 63;
  int rm = (i >> 6) & 511;
  int q  = i >> 15;
  int t  = q * CHUNK + (rm >> 4);
  int m  = rm & 15;
  Ucat[i] = u[(long long)b * (TLEN * MCTL) + (long long)t * MCTL + m];
}

// y_nat[b, n] = y_obs[b, n] - effect[n, b]
__global__ void ynat_kernel(const float* __restrict__ yobs,
                            const float* __restrict__ eff,
                            float* __restrict__ ynat) {
  int i = blockIdx.x * blockDim.x + threadIdx.x;      // 64*512
  if (i >= BATCH * NDIM) return;
  int b = i >> 9, n = i & (NDIM - 1);
  ynat[i] = yobs[i] - eff[(long long)n * BATCH + b];
}

// Z[b, i, n] = sum_k sigma_phi_M[i, k] * Yk[b, k, n]
// Yk[b,0]=y_nat[b]; Yk[b,k>=1]=y_nat_history[b, 20-k]
__global__ void z_kernel(const float* __restrict__ sigM,
                         const float* __restrict__ ynat,
                         const float* __restrict__ ynh,
                         float* __restrict__ Z) {
  int i = blockIdx.x * blockDim.x + threadIdx.x;      // 64*10*512
  if (i >= BATCH * HSP * NDIM) return;
  int n  = i & (NDIM - 1);
  int bi = i >> 9;                                    // b*10 + i
  int ii = bi % HSP;
  int b  = bi / HSP;
  float s = sigM[ii * MSP + 0] * ynat[(long long)b * NDIM + n];
  for (int k = 1; k < MSP; ++k)
    s += sigM[ii * MSP + k] *
         ynh[((long long)b * (2 * MSP) + (2 * MSP - k)) * NDIM + n];
  Z[i] = s;
}

// u[b, c] = bias[c] - K[c,:]·y_obs[b,:] + M0s[c,:]·y_nat[b,:]
//           + sum_i R[c,i,:]·Z[b,i,:]
__global__ __launch_bounds__(256) void final_kernel(
    const float* __restrict__ Kmat, const float* __restrict__ bias,
    const float* __restrict__ M0s, const float* __restrict__ R,
    const float* __restrict__ Z, const float* __restrict__ ynat,
    const float* __restrict__ yobs, float* __restrict__ out) {
  int wv = blockIdx.x * (blockDim.x >> 5) + (threadIdx.x >> 5);
  if (wv >= BATCH * MCTL) return;
  int lane = threadIdx.x & 31;
  int b = wv >> 4, c = wv & 15;
  float s = 0.f;
  for (int n = lane; n < NDIM; n += 32)
    s += M0s[c * NDIM + n] * ynat[(long long)b * NDIM + n] -
         Kmat[c * NDIM + n] * yobs[(long long)b * NDIM + n];
  for (int i = 0; i < HSP; ++i)
    for (int n = lane; n < NDIM; n += 32)
      s += R[((long long)c * HSP + i) * NDIM + n] *
           Z[((long long)b * HSP + i) * NDIM + n];
  for (int off = 16; off > 0; off >>= 1) s += __shfl_down(s, off, 32);
  if (lane == 0) out[b * MCTL + c] = s + bias[c];
}

// ---------------------------------------------------------------------------
// Host-side launch helper
// ---------------------------------------------------------------------------
template <int NT, int KC, int LDB>
static inline void gemmL(hipStream_t st,
                         const float* A, int lda, long long sA,
                         const float* B, long long sB,
                         const float* C, int ldc, long long sC,
                         float* D, int ldd, long long sD,
                         int M, int N, int K, int batches) {
  int waves  = (M / 16) * (N / (16 * NT));
  int blocks = (waves + 7) / 8;
  wmma_gemm<NT, KC, LDB><<<dim3(blocks, batches), dim3(256), 0, st>>>(
      A, lda, sA, B, sB, C, ldc, sC, D, ldd, sD, M, N, K);
}

extern "C" void kernel_launch(void* const* d_in, const int* in_sizes, int n_in,
                              void* d_out, int out_size, void* d_ws, size_t ws_size,
                              hipStream_t stream) {
  (void)in_sizes; (void)n_in; (void)out_size; (void)ws_size;
  const float* Amat = (const float*)d_in[0];   // 512x512
  const float* Bin  = (const float*)d_in[1];   // 512x16
  const float* Cmat = (const float*)d_in[2];   // 512x512
  const float* Kmat = (const float*)d_in[3];   // 16x512
  const float* bias = (const float*)d_in[4];   // 16
  const float* M0   = (const float*)d_in[5];   // 16x10x512
  const float* Mten = (const float*)d_in[6];   // 16x10x10x512
  const float* spm  = (const float*)d_in[7];   // 10x10
  const float* spM  = (const float*)d_in[8];   // 10x10
  const float* uhr  = (const float*)d_in[9];   // 64x512x16
  const float* ynh  = (const float*)d_in[10];  // 64x20x512
  const float* yobs = (const float*)d_in[11];  // 64x512
  float* out = (float*)d_out;                  // 64x16

  // -------- workspace carve-up (floats) --------
  float* ws   = (float*)d_ws;
  float* Wcat = ws;                              // 512 x 512  (W_r in cols r*16..)
  float* T0   = Wcat + NDIM * TLEN;              // 512 x 512
  float* T1   = T0 + NDIM * NDIM;                // 512 x 512
  float* Ucat = T1 + NDIM * NDIM;                // 16 x 512 x 64
  float* P    = Ucat + NQ * TLEN * BATCH;        // 16 x 512 x 64
  float* S0   = P + NQ * NDIM * BATCH;           // 512 x 64
  float* S1   = S0 + NDIM * BATCH;               // 512 x 64
  float* Eff  = S1 + NDIM * BATCH;               // 512 x 64
  float* Ynat = Eff + NDIM * BATCH;              // 64 x 512
  float* Sm   = Ynat + BATCH * NDIM;             // 16 (10 used)
  float* M0s  = Sm + 16;                         // 16 x 512
  float* Rb   = M0s + MCTL * NDIM;               // 16 x 10 x 512
  float* Zb   = Rb + MCTL * HSP * NDIM;          // 64 x 10 x 512

  const long long PS = (long long)NDIM * BATCH;  // 32768

  // -------- small precomputes --------
  sm_kernel<<<1, 32, 0, stream>>>(spm, Sm);
  m0s_kernel<<<(MCTL * NDIM) / 256, 256, 0, stream>>>(M0, Sm, M0s);
  r_kernel<<<(MCTL * HSP * NDIM) / 256, 256, 0, stream>>>(Mten, Sm, Rb);
  copyB_kernel<<<(NDIM * MCTL) / 256, 256, 0, stream>>>(Bin, Wcat);
  packU_kernel<<<(NQ * TLEN * BATCH) / 256, 256, 0, stream>>>(uhr, Ucat);

  // -------- stage A: W_r = A^r B, r = 1..31 (serial 512x16x512 GEMMs) -----
  // Latency-critical: NT=1 (N=16), KC=4 independent K-chains per wave.
  for (int r = 1; r < CHUNK; ++r)
    gemmL<1, 4, TLEN>(stream, Amat, NDIM, 0, Wcat + (r - 1) * MCTL, 0,
                      nullptr, 0, 0, Wcat + r * MCTL, TLEN, 0,
                      NDIM, MCTL, NDIM, 1);

  // -------- stage B: A^32 via 5 squarings (512^3 WMMA GEMMs) --------------
  gemmL<4, 2, NDIM>(stream, Amat, NDIM, 0, Amat, 0, nullptr, 0, 0, T0, NDIM, 0,
                    NDIM, NDIM, NDIM, 1);                            // A^2
  gemmL<4, 2, NDIM>(stream, T0, NDIM, 0, T0, 0, nullptr, 0, 0, T1, NDIM, 0,
                    NDIM, NDIM, NDIM, 1);                            // A^4
  gemmL<4, 2, NDIM>(stream, T1, NDIM, 0, T1, 0, nullptr, 0, 0, T0, NDIM, 0,
                    NDIM, NDIM, NDIM, 1);                            // A^8
  gemmL<4, 2, NDIM>(stream, T0, NDIM, 0, T0, 0, nullptr, 0, 0, T1, NDIM, 0,
                    NDIM, NDIM, NDIM, 1);                            // A^16
  gemmL<4, 2, NDIM>(stream, T1, NDIM, 0, T1, 0, nullptr, 0, 0, T0, NDIM, 0,
                    NDIM, NDIM, NDIM, 1);                            // A^32 in T0

  // -------- stage C: P_q = Wcat @ Ucat_q  (batched over q) ----------------
  gemmL<4, 2, BATCH>(stream, Wcat, TLEN, 0, Ucat, PS, nullptr, 0, 0,
                     P, BATCH, PS, NDIM, BATCH, TLEN, NQ);

  // -------- stage D: Horner  s = P_0 + A32*(P_1 + A32*(...)) --------------
  const float* sin_ = P + 15 * PS;
  float* sout = S0;
  for (int q = 14; q >= 0; --q) {
    gemmL<4, 2, BATCH>(stream, T0, NDIM, 0, sin_, 0,
                       P + (long long)q * PS, BATCH, 0,
                       sout, BATCH, 0, NDIM, BATCH, NDIM, 1);
    sin_ = sout;
    sout = (sout == S0) ? S1 : S0;
  }

  // -------- stage E: effect = C @ s ; y_nat ; control ---------------------
  gemmL<4, 2, BATCH>(stream, Cmat, NDIM, 0, sin_, 0, nullptr, 0, 0,
                     Eff, BATCH, 0, NDIM, BATCH, NDIM, 1);
  ynat_kernel<<<(BATCH * NDIM) / 256, 256, 0, stream>>>(yobs, Eff, Ynat);
  z_kernel<<<(BATCH * HSP * NDIM) / 256, 256, 0, stream>>>(spM, Ynat, ynh, Zb);
  final_kernel<<<(BATCH * MCTL) / 8, 256, 0, stream>>>(Kmat, bias, M0s, Rb, Zb,
                                                       Ynat, yobs, out);
}